// MultiHeadAttentionQuantum_65481071401854
// MI455X (gfx1250) — compile-verified
//
#include <hip/hip_runtime.h>

typedef float v2f __attribute__((ext_vector_type(2)));
typedef float v8f __attribute__((ext_vector_type(8)));
typedef int   vi4 __attribute__((vector_size(16)));

#define S_LEN   2048
#define E_DIM   1024
#define H_HEADS 128
#define DK      8

#define AS1 __attribute__((address_space(1)))
#define AS3 __attribute__((address_space(3)))

// ---- async global -> LDS copy (16B per lane), ASYNCcnt-tracked ----
__device__ __forceinline__ void async_copy_b128(void* ldsp, const void* gp) {
#if __has_builtin(__builtin_amdgcn_global_load_async_to_lds_b128)
    __builtin_amdgcn_global_load_async_to_lds_b128(
        (AS1 vi4*)(uintptr_t)gp,                      // global src (AS1), v4i element
        (AS3 vi4*)(uint32_t)(uintptr_t)ldsp,          // LDS dst: low 32 bits = LDS offset
        0, 0);
#else
    const float4 v = *(const float4*)gp;
    float* d = (float*)ldsp;
    d[0] = v.x; d[1] = v.y; d[2] = v.z; d[3] = v.w;
#endif
}

__device__ __forceinline__ void wait_async_le4() {
#if __has_builtin(__builtin_amdgcn_s_wait_asynccnt)
    __builtin_amdgcn_s_wait_asynccnt(4);
#else
    asm volatile("s_wait_asynccnt 0x4" ::: "memory");
#endif
}
__device__ __forceinline__ void wait_async_le0() {
#if __has_builtin(__builtin_amdgcn_s_wait_asynccnt)
    __builtin_amdgcn_s_wait_asynccnt(0);
#else
    asm volatile("s_wait_asynccnt 0x0" ::: "memory");
#endif
}

// -------------------------------------------------------------------------
// Kernel 1: per (b,s) quantum-head attention.
//   q[h][d] = cos(x[b,s,h*8+d] + theta[d])            (LDS, 128x8)
//   S = Q.Q^T / sqrt(8)  via v_wmma_f32_16x16x4_f32    (8 waves x 16 rows)
//   P = softmax_rows(S)   (register softmax, 16-lane shfl_xor butterflies)
//   O = P.Q  (K=128, N=8) via VALU FMA + butterfly reduce
//   scatter O into out_pre[b, 16h + s/128, (s%128)*8 + d]
// -------------------------------------------------------------------------
__global__ __launch_bounds__(256) void qattn_kernel(const float* __restrict__ x,
                                                    const float* __restrict__ theta,
                                                    float* __restrict__ out_pre) {
    __shared__ float q[H_HEADS][DK + 1];   // pad to 9 floats to break bank conflicts

    const int bs = blockIdx.x;             // 0 .. B*S-1
    const int b  = bs >> 11;               // / 2048
    const int s  = bs & (S_LEN - 1);
    const int tid = threadIdx.x;

    // ---- stage q into LDS: 256 threads x 4 floats = 1024 = E ----
    {
        const float4 xv = *(const float4*)(x + (size_t)bs * E_DIM + tid * 4);
        const int e = tid * 4;
        const int h = e >> 3;
        const int d = e & 7;               // 0 or 4: float4 never crosses a head
        q[h][d + 0] = __cosf(xv.x + theta[d + 0]);
        q[h][d + 1] = __cosf(xv.y + theta[d + 1]);
        q[h][d + 2] = __cosf(xv.z + theta[d + 2]);
        q[h][d + 3] = __cosf(xv.w + theta[d + 3]);
    }
    __syncthreads();

    const int lane = tid & 31;
    const int wave = tid >> 5;             // 8 waves; wave w owns score rows 16w..16w+15
    const int half = lane >> 4;            // WMMA lane-half
    const int lm   = lane & 15;

    // A operand (16x4 f32 layout): VGPR0 = K(2*half), VGPR1 = K(2*half+1)
    const int mrow = wave * 16 + lm;
    v2f a0, a1;
    a0.x = q[mrow][2 * half + 0];
    a0.y = q[mrow][2 * half + 1];
    a1.x = q[mrow][4 + 2 * half + 0];
    a1.y = q[mrow][4 + 2 * half + 1];

    // ---- scores: 8 tiles of 16x16, K=8 (two wmma per tile) ----
    v8f acc[8];
    #pragma unroll
    for (int t = 0; t < 8; ++t) {
        const int col = 16 * t + lm;
        v2f b0, b1;
        b0.x = q[col][2 * half + 0];
        b0.y = q[col][2 * half + 1];
        b1.x = q[col][4 + 2 * half + 0];
        b1.y = q[col][4 + 2 * half + 1];
        v8f c = {0.f, 0.f, 0.f, 0.f, 0.f, 0.f, 0.f, 0.f};
        c = __builtin_amdgcn_wmma_f32_16x16x4_f32(false, a0, false, b0, (short)0, c, false, false);
        c = __builtin_amdgcn_wmma_f32_16x16x4_f32(false, a1, false, b1, (short)0, c, false, false);
        acc[t] = c;
    }

    const float scale = 0.35355339059327373f;   // 1/sqrt(8)
    #pragma unroll
    for (int t = 0; t < 8; ++t)
        #pragma unroll
        for (int r = 0; r < 8; ++r)
            acc[t][r] *= scale;

    // ---- row softmax: C/D layout => row m = r + 8*half (fixed per VGPR),
    //      columns n = lm across a 16-lane group; reduce with xor masks 1,2,4,8 ----
    #pragma unroll
    for (int r = 0; r < 8; ++r) {
        float mx = acc[0][r];
        #pragma unroll
        for (int t = 1; t < 8; ++t) mx = fmaxf(mx, acc[t][r]);
        #pragma unroll
        for (int m = 1; m < 16; m <<= 1) mx = fmaxf(mx, __shfl_xor(mx, m, 32));
        float sum = 0.f;
        #pragma unroll
        for (int t = 0; t < 8; ++t) {
            const float ev = __expf(acc[t][r] - mx);
            acc[t][r] = ev;
            sum += ev;
        }
        #pragma unroll
        for (int m = 1; m < 16; m <<= 1) sum += __shfl_xor(sum, m, 32);
        const float inv = 1.0f / sum;
        #pragma unroll
        for (int t = 0; t < 8; ++t) acc[t][r] *= inv;
    }

    // ---- O = P.Q (per row: dot over 128 cols, N=8) + reinterleaved scatter ----
    #pragma unroll
    for (int r = 0; r < 8; ++r) {
        float o[8] = {0.f, 0.f, 0.f, 0.f, 0.f, 0.f, 0.f, 0.f};
        #pragma unroll
        for (int t = 0; t < 8; ++t) {
            const int col = 16 * t + lm;
            const float pv = acc[t][r];
            #pragma unroll
            for (int d = 0; d < 8; ++d) o[d] = fmaf(pv, q[col][d], o[d]);
        }
        #pragma unroll
        for (int m = 1; m < 16; m <<= 1)
            #pragma unroll
            for (int d = 0; d < 8; ++d) o[d] += __shfl_xor(o[d], m, 32);

        if (lm == 0) {                      // lanes 0 and 16 each own one row
            const int h      = 16 * wave + r + 8 * half;
            const int sprime = 16 * h + (s >> 7);
            const int ebase  = (s & 127) * 8;
            float* dst = out_pre + ((size_t)(b * S_LEN + sprime)) * E_DIM + ebase;
            float4 v0 = {o[0], o[1], o[2], o[3]};
            float4 v1 = {o[4], o[5], o[6], o[7]};
            *(float4*)(dst + 0) = v0;
            *(float4*)(dst + 4) = v1;
        }
    }
}

// -------------------------------------------------------------------------
// Kernel 2: C(Mx1024) = A(Mx1024) . W^T + bias, fp32 WMMA 16x16x4.
// 128x128 output tile per workgroup, BK=16, 8 waves x (8 tiles x 4 ksteps).
// Double-buffered async global->LDS staging (ASYNCcnt pipelined).
// -------------------------------------------------------------------------
#define PAD 20   // 80B row stride: keeps b128 LDS writes 16B-aligned, conflict-free reads

__global__ __launch_bounds__(256) void gemm_bias_kernel(const float* __restrict__ A,
                                                        const float* __restrict__ W,
                                                        const float* __restrict__ bias,
                                                        float* __restrict__ out) {
    __shared__ float As[2][128][PAD];
    __shared__ float Bs[2][128][PAD];

    const int tid  = threadIdx.x;
    const int lane = tid & 31;
    const int wave = tid >> 5;
    const int half = lane >> 4;
    const int lm   = lane & 15;

    const int rowBase = blockIdx.y * 128;
    const int colBase = blockIdx.x * 128;

    // Each thread stages 2 float4 per matrix per chunk (4 async b128 ops).
    auto load_chunk = [&](int buf, int k0) {
        #pragma unroll
        for (int i = 0; i < 2; ++i) {
            const int id2 = tid + i * 256;
            const int row = id2 >> 2;
            const int kk  = (id2 & 3) * 4;
            async_copy_b128(&As[buf][row][kk],
                            A + (size_t)(rowBase + row) * E_DIM + k0 + kk);
            async_copy_b128(&Bs[buf][row][kk],
                            W + (size_t)(colBase + row) * E_DIM + k0 + kk);
        }
    };

    v8f acc[8];
    #pragma unroll
    for (int t = 0; t < 8; ++t) {
        v8f z = {0.f, 0.f, 0.f, 0.f, 0.f, 0.f, 0.f, 0.f};
        acc[t] = z;
    }

    load_chunk(0, 0);                       // prologue: 4 async ops in flight
    int buf = 0;

    for (int k0 = 0; k0 < E_DIM; k0 += 16) {
        if (k0 + 16 < E_DIM) {
            load_chunk(buf ^ 1, k0 + 16);   // prefetch next chunk (4 more ops)
            wait_async_le4();               // chunk k0's 4 ops complete (in-order)
        } else {
            wait_async_le0();
        }
        __syncthreads();

        #pragma unroll
        for (int kc = 0; kc < 16; kc += 4) {
            v2f a;
            a.x = As[buf][16 * wave + lm][kc + 2 * half + 0];
            a.y = As[buf][16 * wave + lm][kc + 2 * half + 1];
            #pragma unroll
            for (int t = 0; t < 8; ++t) {
                v2f bm;
                bm.x = Bs[buf][16 * t + lm][kc + 2 * half + 0];
                bm.y = Bs[buf][16 * t + lm][kc + 2 * half + 1];
                acc[t] = __builtin_amdgcn_wmma_f32_16x16x4_f32(false, a, false, bm,
                                                               (short)0, acc[t], false, false);
            }
        }
        __syncthreads();
        buf ^= 1;
    }

    // ---- bias + store (C/D layout: row = r + 8*half, col = 16t + lm) ----
    #pragma unroll
    for (int t = 0; t < 8; ++t) {
        const int j  = colBase + 16 * t + lm;
        const float bv = bias[j];
        #pragma unroll
        for (int r = 0; r < 8; ++r) {
            const int rowg = rowBase + 16 * wave + r + 8 * half;
            out[(size_t)rowg * E_DIM + j] = acc[t][r] + bv;
        }
    }
}

extern "C" void kernel_launch(void* const* d_in, const int* in_sizes, int n_in,
                              void* d_out, int out_size, void* d_ws, size_t ws_size,
                              hipStream_t stream) {
    const float* x     = (const float*)d_in[0];
    const float* theta = (const float*)d_in[1];
    const float* W     = (const float*)d_in[2];
    const float* bias  = (const float*)d_in[3];
    float* out         = (float*)d_out;

    const int B = in_sizes[0] / (S_LEN * E_DIM);   // = 4
    const int M = B * S_LEN;                       // = 8192

    float* out_pre = (float*)d_ws;                 // M x 1024 f32 staging (32 MB, L2-resident)

    qattn_kernel<<<B * S_LEN, 256, 0, stream>>>(x, theta, out_pre);

    dim3 grid(E_DIM / 128, M / 128);               // (8, 64)
    gemm_bias_kernel<<<grid, dim3(256), 0, stream>>>(out_pre, W, bias, out);
}